// PolMod_1889785610441
// MI455X (gfx1250) — compile-verified
//
#include <hip/hip_runtime.h>
#include <hip/hip_bf16.h>
#include <math.h>

typedef __attribute__((ext_vector_type(2))) float v2f;
typedef __attribute__((ext_vector_type(8))) float v8f;

// ---------------------------------------------------------------------------
// Kernel 1: ch2_def = [M1flat; M2flat] (2x4) @ basis (4xN) via V_WMMA_F32_16X16X4_F32
// One wave handles 16 points. A-layout (f32 16x4): lane m -> K=0,1 ; lane m+16 -> K=2,3.
// B-layout (f32 4x16): lane m -> K=0,1 (cols) ; lane m+16 -> K=2,3.
// D rows M=0,1 land in VGPR0/1 of lanes 0..15 -> store float2 per point.
// ---------------------------------------------------------------------------
__global__ void __launch_bounds__(256) poly_wmma_kernel(
    const float* __restrict__ ch2, const float* __restrict__ M1,
    const float* __restrict__ M2, float2* __restrict__ ch2def, int N)
{
    const int lane = threadIdx.x & 31;
    const int wave = threadIdx.x >> 5;
    const int m    = lane & 15;
    const int half = lane >> 4;
    const int base = (blockIdx.x * 8 + wave) * 16;
    const int p    = base + m;

    if (base + 16 <= N) {
        // Wave-uniform branch: EXEC is all ones here (required for WMMA).
        const float x1 = ch2[p];
        const float x2 = ch2[p + N];

        v2f a; a.x = 0.0f; a.y = 0.0f;
        if (m == 0)      { a.x = half ? M1[2] : M1[0]; a.y = half ? M1[3] : M1[1]; }
        else if (m == 1) { a.x = half ? M2[2] : M2[0]; a.y = half ? M2[3] : M2[1]; }

        v2f b;
        b.x = half ? x1        : 1.0f;  // K=2 (x1)     : K=0 (1)
        b.y = half ? x1 * x2   : x2;    // K=3 (x1*x2)  : K=1 (x2)

        v8f c = {};
        c = __builtin_amdgcn_wmma_f32_16x16x4_f32(
                /*neg_a=*/false, a, /*neg_b=*/false, b,
                /*c_mod=*/(short)0, c, /*reuse_a=*/false, /*reuse_b=*/false);

        if (half == 0) {
            float2 r; r.x = c[0]; r.y = c[1];   // rows M=0, M=1 of this column
            ch2def[p] = r;
        }
    } else {
        // Scalar tail (N % 128 != 0 case); unused for N = 2,000,000.
        if (half == 0 && p < N) {
            const float x1 = ch2[p];
            const float x2 = ch2[p + N];
            const float b1 = x2, b2 = x1, b3 = x1 * x2;
            float2 r;
            r.x = M1[0] + M1[1] * b1 + M1[2] * b2 + M1[3] * b3;
            r.y = M2[0] + M2[1] * b1 + M2[2] * b2 + M2[3] * b3;
            ch2def[p] = r;
        }
    }
}

// ---------------------------------------------------------------------------
// Kernel 0: zero the double accumulator (d_ws is poisoned by the harness).
// ---------------------------------------------------------------------------
__global__ void init_acc_kernel(double* acc)
{
    if (threadIdx.x == 0 && blockIdx.x == 0) *acc = 0.0;
}

// ---------------------------------------------------------------------------
// Kernel 2: per node i, sum w over its 4 edges (L2-resident gathers into
// ch2_def), take log, block-reduce, atomicAdd into the double accumulator.
// ---------------------------------------------------------------------------
__global__ void __launch_bounds__(256) entropy_kernel(
    const float*  __restrict__ ch1,
    const float2* __restrict__ ch2def,
    const int*    __restrict__ nn_ch1,
    const int*    __restrict__ nn_ch2,
    double* acc, int N)
{
    const int i = blockIdx.x * 256 + threadIdx.x;
    float lg = 0.0f;
    if (i < N) {
        const int e  = i * 4;
        const int ii = nn_ch1[e];          // == i for the reference's edge list
        const float mi0 = ch1[ii];
        const float mi1 = ch1[ii + N];
        const int4 idx = *(const int4*)(nn_ch2 + e);   // 16B aligned, coalesced

        float wsum = 0.0f;
        {
            float2 mj = ch2def[idx.x];
            float dx = mi0 - mj.x, dy = mi1 - mj.y;
            wsum += __expf(-(1.0f + (dx * dx + dy * dy) * (1.0f / 1.5f)));
        }
        {
            float2 mj = ch2def[idx.y];
            float dx = mi0 - mj.x, dy = mi1 - mj.y;
            wsum += __expf(-(1.0f + (dx * dx + dy * dy) * (1.0f / 1.5f)));
        }
        {
            float2 mj = ch2def[idx.z];
            float dx = mi0 - mj.x, dy = mi1 - mj.y;
            wsum += __expf(-(1.0f + (dx * dx + dy * dy) * (1.0f / 1.5f)));
        }
        {
            float2 mj = ch2def[idx.w];
            float dx = mi0 - mj.x, dy = mi1 - mj.y;
            wsum += __expf(-(1.0f + (dx * dx + dy * dy) * (1.0f / 1.5f)));
        }
        lg = __logf(wsum);
    }

    // wave32 reduction
    #pragma unroll
    for (int off = 16; off > 0; off >>= 1)
        lg += __shfl_down(lg, off, 32);

    __shared__ float partial[8];
    const int lane = threadIdx.x & 31;
    const int wave = threadIdx.x >> 5;
    if (lane == 0) partial[wave] = lg;
    __syncthreads();

    if (wave == 0) {
        float v = (lane < 8) ? partial[lane] : 0.0f;
        #pragma unroll
        for (int off = 4; off > 0; off >>= 1)
            v += __shfl_down(v, off, 32);
        if (lane == 0) atomicAdd(acc, (double)v);  // global_atomic_add_f64
    }
}

// ---------------------------------------------------------------------------
// Kernel 3: out = log(N) - acc / N
// ---------------------------------------------------------------------------
__global__ void finalize_kernel(const double* acc, float* out, int N)
{
    if (threadIdx.x == 0 && blockIdx.x == 0)
        out[0] = logf((float)N) - (float)(*acc / (double)N);
}

extern "C" void kernel_launch(void* const* d_in, const int* in_sizes, int n_in,
                              void* d_out, int out_size, void* d_ws, size_t ws_size,
                              hipStream_t stream)
{
    const float* ch1 = (const float*)d_in[0];   // (2, N)
    const float* ch2 = (const float*)d_in[1];   // (2, N)
    const float* M1  = (const float*)d_in[2];   // (2, 2)
    const float* M2  = (const float*)d_in[3];   // (2, 2)
    const int*   nn1 = (const int*)d_in[4];     // (N*4,)
    const int*   nn2 = (const int*)d_in[5];     // (N*4,)
    const int N = in_sizes[0] / 2;

    double* acc    = (double*)d_ws;                       // 16B-aligned slot
    float2* ch2def = (float2*)((char*)d_ws + 16);         // N * 8 bytes

    init_acc_kernel<<<1, 1, 0, stream>>>(acc);

    const int tiles = (N + 127) / 128;                    // 128 points per 256-thread block
    poly_wmma_kernel<<<tiles, 256, 0, stream>>>(ch2, M1, M2, ch2def, N);

    entropy_kernel<<<(N + 255) / 256, 256, 0, stream>>>(ch1, ch2def, nn1, nn2, acc, N);

    finalize_kernel<<<1, 1, 0, stream>>>(acc, (float*)d_out, N);
}